// GCN_35459249995962
// MI455X (gfx1250) — compile-verified
//
#include <hip/hip_runtime.h>
#include <hip/hip_bf16.h>
#include <math.h>

typedef __attribute__((ext_vector_type(2))) float v2f;
typedef __attribute__((ext_vector_type(8))) float v8f;

#define N_NODES 50000
#define N_EDGES 800000
#define CH 128
#define N_LAYERS 3
#define N_GRAPHS 128
#define FEAT 512            // 128 + 3*128
#define OUT_CH 10
#define BN_EPS 1e-5f

// ---------------------------------------------------------------------------
// GEMM: Out[N,128] = A[N,128] @ W[128,128], fp32 WMMA 16x16x4.
// One wave per 16x16 tile. 8 waves/block -> 16 rows x 128 cols per block.
// A layout (ISA 7.12.2, 32-bit A 16x4): lanes0-15 hold K=0(K=1 in v1),
// lanes16-31 hold K=2,3. B mirrors (rows split across lane halves).
// D layout: VGPR r -> rows r (lanes0-15) and r+8 (lanes16-31), N = lane&15.
// ---------------------------------------------------------------------------
__global__ void __launch_bounds__(256)
gcn_gemm128_wmma(const float* __restrict__ A, const float* __restrict__ W,
                 float* __restrict__ Out, int N) {
    const int lane = threadIdx.x & 31;
    const int wave = threadIdx.x >> 5;          // 0..7 -> column tile
    const int row0 = blockIdx.x << 4;           // 16 rows per block
    const int col0 = wave << 4;
    const int m    = lane & 15;
    const int kh   = (lane >> 4) << 1;          // 0 or 2 (lane half selects K pair)

    int arow = row0 + m;
    if (arow > N - 1) arow = N - 1;             // clamp (no EXEC divergence)
    const float* __restrict__ ap = A + (size_t)arow * CH;

    v8f acc = {};
#pragma unroll
    for (int k0 = 0; k0 < CH; k0 += 4) {
        v2f a, b;
        a.x = ap[k0 + kh];
        a.y = ap[k0 + kh + 1];
        b.x = W[(size_t)(k0 + kh)     * CH + col0 + m];
        b.y = W[(size_t)(k0 + kh + 1) * CH + col0 + m];
        acc = __builtin_amdgcn_wmma_f32_16x16x4_f32(
            /*neg_a=*/false, a, /*neg_b=*/false, b,
            /*c_mod=*/(short)0, acc, /*reuse_a=*/false, /*reuse_b=*/false);
    }

    const int rbase = (lane >> 4) << 3;         // lanes16-31 -> rows +8
#pragma unroll
    for (int r = 0; r < 8; ++r) {
        const int orow = row0 + r + rbase;
        if (orow < N) Out[(size_t)orow * CH + col0 + m] = acc[r];
    }
}

// ---------------------------------------------------------------------------
// Degree with self-loops: deg[v] = 1 + #incoming edges; then dinv = rsqrt.
// ---------------------------------------------------------------------------
__global__ void deg_init(float* __restrict__ deg, int N) {
    int v = blockIdx.x * blockDim.x + threadIdx.x;
    if (v < N) deg[v] = 1.0f;
}
__global__ void deg_edges(const int* __restrict__ dst, float* __restrict__ deg, int E) {
    int e = blockIdx.x * blockDim.x + threadIdx.x;
    if (e < E) atomicAdd(&deg[dst[e]], 1.0f);
}
__global__ void deg_rsqrt(float* __restrict__ deg, int N) {
    int v = blockIdx.x * blockDim.x + threadIdx.x;
    if (v < N) deg[v] = rsqrtf(deg[v]);
}

// ---------------------------------------------------------------------------
// Self-loop term (plain store -> doubles as zero-init of the accumulator):
// hagg[v][:] = hlin[v][:] * dinv[v]^2
// ---------------------------------------------------------------------------
__global__ void self_loop(const float* __restrict__ hlin, const float* __restrict__ dinv,
                          float* __restrict__ hagg, int N) {
    int idx = blockIdx.x * blockDim.x + threadIdx.x;   // N*32 threads, 4 ch each
    int v = idx >> 5;
    if (v >= N) return;
    int c = (idx & 31) << 2;
    float s = dinv[v]; s *= s;
    float4 h = *reinterpret_cast<const float4*>(hlin + (size_t)v * CH + c);
    float4 o; o.x = h.x * s; o.y = h.y * s; o.z = h.z * s; o.w = h.w * s;
    *reinterpret_cast<float4*>(hagg + (size_t)v * CH + c) = o;
}

// ---------------------------------------------------------------------------
// Edge scatter: hagg[dst] += hlin[src] * dinv[src]*dinv[dst]. 32 threads/edge.
// ---------------------------------------------------------------------------
__global__ void edge_scatter(const float* __restrict__ hlin, const int* __restrict__ src,
                             const int* __restrict__ dst, const float* __restrict__ dinv,
                             float* __restrict__ hagg, int E) {
    int idx = blockIdx.x * blockDim.x + threadIdx.x;   // E*32 threads
    int e = idx >> 5;
    if (e >= E) return;
    int c = (idx & 31) << 2;
    int s = src[e], d = dst[e];
    float nrm = dinv[s] * dinv[d];
    float4 h = *reinterpret_cast<const float4*>(hlin + (size_t)s * CH + c);
    float* o = hagg + (size_t)d * CH + c;
    atomicAdd(o + 0, h.x * nrm);
    atomicAdd(o + 1, h.y * nrm);
    atomicAdd(o + 2, h.z * nrm);
    atomicAdd(o + 3, h.w * nrm);
}

// ---------------------------------------------------------------------------
// Fused bias + BatchNorm(eval) + ReLU -> persistent layer buffer.
// ---------------------------------------------------------------------------
__global__ void bias_bn_relu(const float* __restrict__ hagg, const float* __restrict__ b,
                             const float* __restrict__ rm, const float* __restrict__ rv,
                             const float* __restrict__ gm, const float* __restrict__ bt,
                             float* __restrict__ outl, int N) {
    int idx = blockIdx.x * blockDim.x + threadIdx.x;   // N*32 threads, 4 ch each
    int v = idx >> 5;
    if (v >= N) return;
    int c = (idx & 31) << 2;
    float4 h = *reinterpret_cast<const float4*>(hagg + (size_t)v * CH + c);
    float4 o;
    o.x = fmaxf((h.x + b[c+0] - rm[c+0]) * rsqrtf(rv[c+0] + BN_EPS) * gm[c+0] + bt[c+0], 0.0f);
    o.y = fmaxf((h.y + b[c+1] - rm[c+1]) * rsqrtf(rv[c+1] + BN_EPS) * gm[c+1] + bt[c+1], 0.0f);
    o.z = fmaxf((h.z + b[c+2] - rm[c+2]) * rsqrtf(rv[c+2] + BN_EPS) * gm[c+2] + bt[c+2], 0.0f);
    o.w = fmaxf((h.w + b[c+3] - rm[c+3]) * rsqrtf(rv[c+3] + BN_EPS) * gm[c+3] + bt[c+3], 0.0f);
    *reinterpret_cast<float4*>(outl + (size_t)v * CH + c) = o;
}

// ---------------------------------------------------------------------------
// Pooling: zero, node counts, atomic feature sums over concat [x,l0,l1,l2].
// ---------------------------------------------------------------------------
__global__ void pool_zero(float* __restrict__ pooled, float* __restrict__ counts) {
    int i = blockIdx.x * blockDim.x + threadIdx.x;
    if (i < N_GRAPHS * FEAT) pooled[i] = 0.0f;
    if (i < N_GRAPHS) counts[i] = 0.0f;
}
__global__ void pool_counts(const int* __restrict__ batch, float* __restrict__ counts, int N) {
    int v = blockIdx.x * blockDim.x + threadIdx.x;
    if (v < N) atomicAdd(&counts[batch[v]], 1.0f);
}
__global__ void pool_sum(const float* __restrict__ x,  const float* __restrict__ l0,
                         const float* __restrict__ l1, const float* __restrict__ l2,
                         const int* __restrict__ batch, float* __restrict__ pooled, int N) {
    int idx = blockIdx.x * blockDim.x + threadIdx.x;   // N*128 threads, 4 of 512 ch each
    int v = idx >> 7;
    if (v >= N) return;
    int q = (idx & 127) << 2;                          // 0..508 within concat feature
    const float* s;
    switch (q >> 7) {
        case 0:  s = x;  break;
        case 1:  s = l0; break;
        case 2:  s = l1; break;
        default: s = l2; break;
    }
    float4 val = *reinterpret_cast<const float4*>(s + (size_t)v * CH + (q & 127));
    float* o = pooled + (size_t)batch[v] * FEAT + q;
    atomicAdd(o + 0, val.x);
    atomicAdd(o + 1, val.y);
    atomicAdd(o + 2, val.z);
    atomicAdd(o + 3, val.w);
}

// ---------------------------------------------------------------------------
// Classifier + log_softmax. One wave (32 threads) per graph.
// ---------------------------------------------------------------------------
__global__ void classifier(const float* __restrict__ pooled, const float* __restrict__ counts,
                           const float* __restrict__ Wc, const float* __restrict__ bc,
                           float* __restrict__ out) {
    __shared__ float logits[OUT_CH];
    int g = blockIdx.x;
    int l = threadIdx.x;
    float inv = 1.0f / fmaxf(counts[g], 1.0f);
    if (l < OUT_CH) {
        float acc = bc[l];
        const float* p = pooled + (size_t)g * FEAT;
        for (int c = 0; c < FEAT; ++c) acc += (p[c] * inv) * Wc[(size_t)c * OUT_CH + l];
        logits[l] = acc;
    }
    __syncthreads();
    float mx = -INFINITY;
    for (int i = 0; i < OUT_CH; ++i) mx = fmaxf(mx, logits[i]);
    float sum = 0.0f;
    for (int i = 0; i < OUT_CH; ++i) sum += expf(logits[i] - mx);
    float lse = mx + logf(sum);
    if (l < OUT_CH) out[(size_t)g * OUT_CH + l] = logits[l] - lse;
}

// ---------------------------------------------------------------------------
extern "C" void kernel_launch(void* const* d_in, const int* in_sizes, int n_in,
                              void* d_out, int out_size, void* d_ws, size_t ws_size,
                              hipStream_t stream) {
    const float* x      = (const float*)d_in[0];
    const int*   eidx   = (const int*)  d_in[1];
    const int*   batch  = (const int*)  d_in[2];
    const float* Ws     = (const float*)d_in[3];
    const float* bs     = (const float*)d_in[4];
    const float* gammas = (const float*)d_in[5];
    const float* betas  = (const float*)d_in[6];
    const float* rms    = (const float*)d_in[7];
    const float* rvs    = (const float*)d_in[8];
    const float* Wc     = (const float*)d_in[9];
    const float* bc     = (const float*)d_in[10];
    float* out = (float*)d_out;

    const int N = in_sizes[0] / CH;        // 50000
    const int E = in_sizes[1] / 2;         // 800000
    const int* src = eidx;
    const int* dst = eidx + E;

    // workspace carve (256B aligned): dinv | hlin | hagg | l0 | l1 | l2 | pooled | counts
    char* ws = (char*)d_ws;
    auto carve = [&](size_t bytes) -> float* {
        float* p = (float*)ws;
        ws += (bytes + 255) & ~(size_t)255;
        return p;
    };
    const size_t nodef = (size_t)N * CH * sizeof(float);
    float* dinv   = carve((size_t)N * sizeof(float));
    float* hlin   = carve(nodef);
    float* hagg   = carve(nodef);
    float* lyr[N_LAYERS];
    for (int i = 0; i < N_LAYERS; ++i) lyr[i] = carve(nodef);
    float* pooled = carve((size_t)N_GRAPHS * FEAT * sizeof(float));
    float* counts = carve((size_t)N_GRAPHS * sizeof(float));

    const int T = 256;
    // 1) degrees / normalization
    deg_init <<<(N + T - 1) / T, T, 0, stream>>>(dinv, N);
    deg_edges<<<(E + T - 1) / T, T, 0, stream>>>(dst, dinv, E);
    deg_rsqrt<<<(N + T - 1) / T, T, 0, stream>>>(dinv, N);

    // 2) GCN layers
    const int gemm_blocks = (N + 15) / 16;
    const int node32      = ((N * 32)  + T - 1) / T;
    const int edge32      = ((E * 32)  + T - 1) / T;
    for (int i = 0; i < N_LAYERS; ++i) {
        const float* hin = (i == 0) ? x : lyr[i - 1];
        gcn_gemm128_wmma<<<gemm_blocks, T, 0, stream>>>(hin, Ws + (size_t)i * CH * CH, hlin, N);
        self_loop       <<<node32, T, 0, stream>>>(hlin, dinv, hagg, N);
        edge_scatter    <<<edge32, T, 0, stream>>>(hlin, src, dst, dinv, hagg, E);
        bias_bn_relu    <<<node32, T, 0, stream>>>(hagg, bs + i * CH, rms + i * CH,
                                                   rvs + i * CH, gammas + i * CH,
                                                   betas + i * CH, lyr[i], N);
    }

    // 3) global mean pool over graphs
    pool_zero  <<<(N_GRAPHS * FEAT + T - 1) / T, T, 0, stream>>>(pooled, counts);
    pool_counts<<<(N + T - 1) / T, T, 0, stream>>>(batch, counts, N);
    pool_sum   <<<((N * CH) + T - 1) / T, T, 0, stream>>>(x, lyr[0], lyr[1], lyr[2],
                                                          batch, pooled, N);

    // 4) classifier + log_softmax
    classifier<<<N_GRAPHS, 32, 0, stream>>>(pooled, counts, Wc, bc, out);
}